// GraphAttentionNetwork_5454608466409
// MI455X (gfx1250) — compile-verified
//
#include <hip/hip_runtime.h>

#define NN   40000
#define EE   640000
#define NEDG (EE + NN)   // edges + self loops
#define FD   128         // feature dim = H*C
#define NH   4           // heads
#define CH   32          // channels per head

typedef __attribute__((ext_vector_type(2))) float v2f;
typedef __attribute__((ext_vector_type(8))) float v8f;

// ---------------- helpers ----------------

__device__ __forceinline__ unsigned f2ord(float f) {
    unsigned u = __float_as_uint(f);
    return (u & 0x80000000u) ? ~u : (u | 0x80000000u);
}
__device__ __forceinline__ float ord2f(unsigned u) {
    return (u & 0x80000000u) ? __uint_as_float(u & 0x7fffffffu)
                             : __uint_as_float(~u);
}
__device__ __forceinline__ void edge_sd(const int* __restrict__ ei, int e, int& s, int& d) {
    if (e < EE) { s = ei[e]; d = ei[EE + e]; }
    else        { s = e - EE; d = e - EE; }     // appended self-loops
}
__device__ __forceinline__ float leaky(float v) { return v >= 0.f ? v : 0.2f * v; }

// ---------------- GEMM: h = x @ W  (fp32 WMMA 16x16x4) ----------------
// grid.x = NN/16 m-tiles, 256 threads = 8 waves; wave w handles n-tile w (16 cols).
__global__ void gat_gemm_wmma(const float* __restrict__ X,
                              const float* __restrict__ W,
                              float* __restrict__ Hout) {
    const int wave = threadIdx.x >> 5;
    const int lane = threadIdx.x & 31;
    const int half = lane >> 4;     // 0: lanes 0-15, 1: lanes 16-31
    const int l16  = lane & 15;
    const int m0   = blockIdx.x * 16;
    const int n0   = wave * 16;

    const float* xrow = X + (size_t)(m0 + l16) * FD;
    v8f acc = {};
#pragma unroll
    for (int k = 0; k < FD; k += 4) {
        // A fragment (16x4): lane<16 holds K=k,k+1 ; lane>=16 holds K=k+2,k+3 (row M=l16)
        v2f a;
        a.x = xrow[k + 2 * half];
        a.y = xrow[k + 2 * half + 1];
        // B fragment (4x16): lane<16 holds rows K=k,k+1 ; lane>=16 rows K=k+2,k+3 (col N=n0+l16)
        v2f b;
        b.x = W[(size_t)(k + 2 * half) * FD + n0 + l16];
        b.y = W[(size_t)(k + 2 * half + 1) * FD + n0 + l16];
        acc = __builtin_amdgcn_wmma_f32_16x16x4_f32(false, a, false, b,
                                                    (short)0, acc, false, false);
    }
    // C/D 16x16 fp32: vgpr r -> row r (lanes 0-15) / row 8+r (lanes 16-31), col = n0+l16
#pragma unroll
    for (int r = 0; r < 8; ++r) {
        int m = m0 + r + 8 * half;
        Hout[(size_t)m * FD + n0 + l16] = acc[r];
    }
}

// ---------------- per-node attention scores ----------------
__global__ void gat_att_scores(const float* __restrict__ Hm,
                               const float* __restrict__ att_src,
                               const float* __restrict__ att_dst,
                               float* __restrict__ as_, float* __restrict__ ad_) {
    int idx = blockIdx.x * blockDim.x + threadIdx.x;   // n*NH + h
    if (idx >= NN * NH) return;
    int n = idx >> 2, hh = idx & 3;
    const float* hp = Hm + (size_t)n * FD + hh * CH;
    float s = 0.f, d = 0.f;
#pragma unroll
    for (int c = 0; c < CH; ++c) {
        float v = hp[c];
        s += v * att_src[hh * CH + c];
        d += v * att_dst[hh * CH + c];
    }
    as_[idx] = s; ad_[idx] = d;
}

// ---------------- per-layer init: out=bias, emax=-inf(key 0), denom=0 ----------------
__global__ void gat_init(float* __restrict__ outb, const float* __restrict__ bias,
                         unsigned* __restrict__ emax, float* __restrict__ denom) {
    int i = blockIdx.x * blockDim.x + threadIdx.x;
    if (i < NN * FD) outb[i] = bias[i & (FD - 1)];
    if (i < NN * NH) { emax[i] = 0u; denom[i] = 0.f; }
}

// ---------------- edge pass A: segment max ----------------
__global__ void gat_edge_max(const int* __restrict__ ei,
                             const float* __restrict__ as_, const float* __restrict__ ad_,
                             unsigned* __restrict__ emax) {
    int e = blockIdx.x * blockDim.x + threadIdx.x;
    if (e >= NEDG) return;
    int s, d; edge_sd(ei, e, s, d);
#pragma unroll
    for (int hh = 0; hh < NH; ++hh) {
        float v = leaky(as_[s * NH + hh] + ad_[d * NH + hh]);
        atomicMax(&emax[d * NH + hh], f2ord(v));
    }
}

// ---------------- edge pass B: segment sum of exp ----------------
__global__ void gat_edge_sum(const int* __restrict__ ei,
                             const float* __restrict__ as_, const float* __restrict__ ad_,
                             const unsigned* __restrict__ emax,
                             float* __restrict__ denom) {
    int e = blockIdx.x * blockDim.x + threadIdx.x;
    if (e >= NEDG) return;
    int s, d; edge_sd(ei, e, s, d);
#pragma unroll
    for (int hh = 0; hh < NH; ++hh) {
        float v = leaky(as_[s * NH + hh] + ad_[d * NH + hh]);
        float w = expf(v - ord2f(emax[d * NH + hh]));
        atomicAdd(&denom[d * NH + hh], w);
    }
}

// ---------------- edge pass C: scatter weighted messages (1 wave per edge) ------------
__global__ void gat_edge_scatter(const int* __restrict__ ei,
                                 const float* __restrict__ Hm,
                                 const float* __restrict__ as_, const float* __restrict__ ad_,
                                 const unsigned* __restrict__ emax,
                                 const float* __restrict__ denom,
                                 float* __restrict__ outb) {
    int gid  = blockIdx.x * blockDim.x + threadIdx.x;
    int e    = gid >> 5;
    int lane = gid & 31;
    if (e >= NEDG) return;
    int s, d; edge_sd(ei, e, s, d);
    float alpha[NH];
#pragma unroll
    for (int hh = 0; hh < NH; ++hh) {
        float v = leaky(as_[s * NH + hh] + ad_[d * NH + hh]);
        float w = expf(v - ord2f(emax[d * NH + hh]));
        alpha[hh] = w / (denom[d * NH + hh] + 1e-16f);
    }
    const float* hs = Hm + (size_t)s * FD;
    float* od = outb + (size_t)d * FD;
#pragma unroll
    for (int j = 0; j < NH; ++j) {
        int c = lane + 32 * j;          // head j, coalesced across lanes
        atomicAdd(&od[c], hs[c] * alpha[j]);
    }
}

// ---------------- BN (eval) + ReLU, in place ----------------
__global__ void gat_bn_relu(float* __restrict__ x,
                            const float* __restrict__ mean, const float* __restrict__ var,
                            const float* __restrict__ gamma, const float* __restrict__ beta) {
    int i = blockIdx.x * blockDim.x + threadIdx.x;
    if (i >= NN * FD) return;
    int f = i & (FD - 1);
    float v = (x[i] - mean[f]) * rsqrtf(var[f] + 1e-5f) * gamma[f] + beta[f];
    x[i] = v > 0.f ? v : 0.f;
}

// ---------------- final projection: y = x @ w_out + b (1 wave per node) ----------------
__global__ void gat_out_proj(const float* __restrict__ x,
                             const float* __restrict__ wout, const float* __restrict__ bout,
                             float* __restrict__ y) {
    int gid  = blockIdx.x * blockDim.x + threadIdx.x;
    int n    = gid >> 5;
    int lane = gid & 31;
    if (n >= NN) return;
    const float* xr = x + (size_t)n * FD;
    float s = 0.f;
#pragma unroll
    for (int j = 0; j < 4; ++j) s += xr[lane * 4 + j] * wout[lane * 4 + j];
#pragma unroll
    for (int off = 16; off > 0; off >>= 1) s += __shfl_xor(s, off, 32);
    if (lane == 0) y[n] = s + bout[0];
}

// ---------------- host side ----------------
extern "C" void kernel_launch(void* const* d_in, const int* in_sizes, int n_in,
                              void* d_out, int out_size, void* d_ws, size_t ws_size,
                              hipStream_t stream) {
    const float* x        = (const float*)d_in[0];
    const int*   ei       = (const int*)  d_in[1];
    const float* W        = (const float*)d_in[2];   // [3,128,128]
    const float* att_src  = (const float*)d_in[3];   // [3,4,32]
    const float* att_dst  = (const float*)d_in[4];
    const float* bias_gat = (const float*)d_in[5];   // [3,128]
    const float* bn_gamma = (const float*)d_in[6];
    const float* bn_beta  = (const float*)d_in[7];
    const float* bn_mean  = (const float*)d_in[8];
    const float* bn_var   = (const float*)d_in[9];
    const float* w_out    = (const float*)d_in[10];  // [128]
    const float* b_out    = (const float*)d_in[11];
    float* y = (float*)d_out;

    // workspace carve-up (floats)
    float*    B0    = (float*)d_ws;                       // NN*FD
    float*    B1    = B0 + (size_t)NN * FD;               // NN*FD
    float*    Hb    = B1 + (size_t)NN * FD;               // NN*FD
    float*    as_   = Hb + (size_t)NN * FD;               // NN*NH
    float*    ad_   = as_ + (size_t)NN * NH;              // NN*NH
    unsigned* emax  = (unsigned*)(ad_ + (size_t)NN * NH); // NN*NH
    float*    denom = (float*)(emax + (size_t)NN * NH);   // NN*NH

    const int TB = 256;
    const int g_nf   = (NN * FD + TB - 1) / TB;
    const int g_nh   = (NN * NH + TB - 1) / TB;
    const int g_e    = (NEDG + TB - 1) / TB;
    const int g_e32  = ((size_t)NEDG * 32 + TB - 1) / TB;
    const int g_n32  = (NN * 32 + TB - 1) / TB;

    for (int l = 0; l < 3; ++l) {
        const float* xin  = (l == 0) ? x : (l == 1 ? B0 : B1);
        float*       outb = (l == 1) ? B1 : B0;   // l0->B0, l1->B1, l2->B0

        gat_gemm_wmma<<<NN / 16, TB, 0, stream>>>(xin, W + (size_t)l * FD * FD, Hb);
        gat_att_scores<<<g_nh, TB, 0, stream>>>(Hb, att_src + l * FD, att_dst + l * FD, as_, ad_);
        gat_init<<<g_nf, TB, 0, stream>>>(outb, bias_gat + l * FD, emax, denom);
        gat_edge_max<<<g_e, TB, 0, stream>>>(ei, as_, ad_, emax);
        gat_edge_sum<<<g_e, TB, 0, stream>>>(ei, as_, ad_, emax, denom);
        gat_edge_scatter<<<g_e32, TB, 0, stream>>>(ei, Hb, as_, ad_, emax, denom, outb);
        gat_bn_relu<<<g_nf, TB, 0, stream>>>(outb, bn_mean + l * FD, bn_var + l * FD,
                                             bn_gamma + l * FD, bn_beta + l * FD);
    }
    gat_out_proj<<<g_n32, TB, 0, stream>>>(B0, w_out, b_out, y);
}